// VectorQuantizer_57269093925494
// MI455X (gfx1250) — compile-verified
//
#include <hip/hip_runtime.h>
#include <hip/hip_bf16.h>
#include <cstdint>

// MI455X / gfx1250, wave32. FP32 WMMA (16x16x4) for the distance matmul,
// async global->LDS staging, branchless streaming softmax reduction.

typedef float v2f __attribute__((ext_vector_type(2)));
typedef float v8f __attribute__((ext_vector_type(8)));
typedef int   gv4i __attribute__((vector_size(16)));   // matches builtin's V4i

#define D 256
#define K 1024
#define MB 32           // x rows per workgroup
#define XS_STRIDE 260   // padded LDS row stride: 16B-aligned rows, 260%64=4
                        // -> conflict-free 16-lane fragment reads

#if defined(__has_builtin)
#if __has_builtin(__builtin_amdgcn_global_load_async_to_lds_b128)
#define HAVE_ASYNC_LDS 1
#endif
#endif

// ws layout (floats): [0..1023] avg-prob column sums, [1024] sum of per-row
// sum(p*logp), [1025] sum of squared quantization error, [1032..2055] ||c_k||^2

__device__ __forceinline__ void merge_stats(float& m, int& i, float& Z, float& W,
                                            float mo, int io, float Zo, float Wo) {
  // keep reference at the smaller distance (larger logit); tie -> smaller index
  if (mo < m || (mo == m && io < i)) {
    float tf;
    tf = m;  m = mo;  mo = tf;
    int ti = i; i = io; io = ti;
    tf = Z;  Z = Zo;  Zo = tf;
    tf = W;  W = Wo;  Wo = tf;
  }
  float dt = -100.0f * (mo - m);          // <= 0
  float s  = __expf(dt);
  W = W + s * (Wo + dt * Zo);
  Z = Z + s * Zo;
}

__global__ __launch_bounds__(256) void vq_init(const float* __restrict__ cb,
                                               float* __restrict__ ws) {
  int g = blockIdx.x * 256 + threadIdx.x;   // 0..1023 (one per codebook row)
  ws[g] = 0.0f;                             // zero column-sum accumulators
  if (g == 0) { ws[1024] = 0.0f; ws[1025] = 0.0f; }
  const float4* p = reinterpret_cast<const float4*>(cb + (size_t)g * D);
  float s = 0.0f;
#pragma unroll 8
  for (int i = 0; i < D / 4; ++i) {
    float4 v = p[i];
    s += v.x * v.x + v.y * v.y + v.z * v.z + v.w * v.w;
  }
  ws[1032 + g] = s;
}

__global__ __launch_bounds__(256) void vq_main(const float* __restrict__ x,
                                               const float* __restrict__ cb,
                                               float* __restrict__ out,
                                               float* __restrict__ ws,
                                               int N) {
  __shared__ float xs[MB * XS_STRIDE];   // staged x tile
  __shared__ float c2s[K];
  __shared__ float colsum[K];
  __shared__ float x2s[MB];
  __shared__ float stm[MB * 4];
  __shared__ int   sti[MB * 4];
  __shared__ float stz[MB * 4];
  __shared__ float stw[MB * 4];
  __shared__ float fm[MB];
  __shared__ int   fi[MB];
  __shared__ float fz[MB];
  __shared__ float plps[MB];
  __shared__ float s_red;

  const int tid = threadIdx.x;
  const int rowbase = blockIdx.x * MB;

  // ---------------- phase 0: stage tiles ----------------
#if HAVE_ASYNC_LDS
  // gfx1250 async DMA: global -> LDS without VGPR round-trip (ASYNCcnt).
  // Low 32 bits of a flat shared address are the LDS offset on this ISA.
#pragma unroll
  for (int it = 0; it < 8; ++it) {
    int f = it * 256 + tid;            // b128 id (2048 = 32 rows * 64)
    int row = f >> 6;
    int c4  = f & 63;
    __builtin_amdgcn_global_load_async_to_lds_b128(
        (__attribute__((address_space(1))) gv4i*)(uintptr_t)
            (x + (size_t)(rowbase + row) * D + c4 * 4),
        (__attribute__((address_space(3))) gv4i*)(unsigned)(uintptr_t)
            (xs + row * XS_STRIDE + c4 * 4),
        0, 0);
  }
#else
#pragma unroll
  for (int it = 0; it < 8; ++it) {
    int f = it * 256 + tid;
    int row = f >> 6;
    int c4  = f & 63;
    float4 v = *reinterpret_cast<const float4*>(
        &x[(size_t)(rowbase + row) * D + c4 * 4]);
    float* p = &xs[row * XS_STRIDE + c4 * 4];
    p[0] = v.x; p[1] = v.y; p[2] = v.z; p[3] = v.w;
  }
#endif
  for (int k = tid; k < K; k += 256) {
    c2s[k] = ws[1032 + k];
    colsum[k] = 0.0f;
  }
  if (tid == 0) s_red = 0.0f;
#if HAVE_ASYNC_LDS
  asm volatile("s_wait_asynccnt 0x0" ::: "memory");
#endif
  __syncthreads();

  if (tid < MB) {
    const float* p = &xs[tid * XS_STRIDE];
    float s = 0.0f;
    for (int c = 0; c < D; ++c) s += p[c] * p[c];
    x2s[tid] = s;
  }
  __syncthreads();

  const int w  = tid >> 5, l = tid & 31;
  const int rt = w & 1,  cg = w >> 1;     // row-tile, column group (0..3)
  const int h  = l >> 4, ln = l & 15;
  const int rslot = rt * 16 + 8 * h;      // local row for acc slot r: rslot + r

  float x2v[8];
#pragma unroll
  for (int r = 0; r < 8; ++r) x2v[r] = x2s[rslot + r];

  // -------- pass 1: distances + streaming softmax stats + argmin --------
  float m_[8], Z_[8], W_[8];
  int bi[8];
#pragma unroll
  for (int r = 0; r < 8; ++r) { m_[r] = 1e30f; Z_[r] = 0.0f; W_[r] = 0.0f; bi[r] = 0; }

  for (int t = 0; t < 16; ++t) {
    const int ct = cg + 4 * t;
    const int col = ct * 16 + ln;
    const float* bp = &cb[(size_t)col * D + 2 * h];
    const float* ap = &xs[(rt * 16 + ln) * XS_STRIDE + 2 * h];
    if (t < 15) __builtin_prefetch(bp + 64 * D, 0, 1);  // next codebook tile
    v8f acc = {};
#pragma unroll 8
    for (int k0 = 0; k0 < D; k0 += 4) {
      v2f a = *reinterpret_cast<const v2f*>(ap + k0);
      v2f b = *reinterpret_cast<const v2f*>(bp + k0);
      acc = __builtin_amdgcn_wmma_f32_16x16x4_f32(false, a, false, b,
                                                  (short)0, acc, false, false);
    }
    const float c2v = c2s[col];
    // branchless flash-softmax update (one of dto/dtn is always 0)
#pragma unroll
    for (int r = 0; r < 8; ++r) {
      float d   = x2v[r] + c2v - 2.0f * acc[r];
      float mn  = fminf(m_[r], d);
      float dto = -100.0f * (m_[r] - mn);   // <= 0; 0 when old min kept
      float dtn = -100.0f * (d - mn);       // <= 0; 0 when d is new min
      float s   = __expf(dto);
      float e   = __expf(dtn);
      W_[r] = s * (W_[r] + dto * Z_[r]) + dtn * e;
      Z_[r] = s * Z_[r] + e;
      bi[r] = (d < m_[r]) ? col : bi[r];
      m_[r] = mn;
    }
  }

  // butterfly all-reduce across the 16 lanes that share h (wave32 shuffles)
#pragma unroll
  for (int mask = 1; mask <= 8; mask <<= 1) {
#pragma unroll
    for (int r = 0; r < 8; ++r) {
      float mo = __shfl_xor(m_[r], mask, 32);
      int   io = __shfl_xor(bi[r], mask, 32);
      float Zo = __shfl_xor(Z_[r], mask, 32);
      float Wo = __shfl_xor(W_[r], mask, 32);
      merge_stats(m_[r], bi[r], Z_[r], W_[r], mo, io, Zo, Wo);
    }
  }
  if (ln == 0) {
#pragma unroll
    for (int r = 0; r < 8; ++r) {
      int rowl = rslot + r;
      stm[rowl * 4 + cg] = m_[r];
      sti[rowl * 4 + cg] = bi[r];
      stz[rowl * 4 + cg] = Z_[r];
      stw[rowl * 4 + cg] = W_[r];
    }
  }
  __syncthreads();

  if (tid < MB) {
    float mm = stm[tid * 4], Zm = stz[tid * 4], Wm = stw[tid * 4];
    int   im = sti[tid * 4];
    for (int g = 1; g < 4; ++g)
      merge_stats(mm, im, Zm, Wm,
                  stm[tid * 4 + g], sti[tid * 4 + g],
                  stz[tid * 4 + g], stw[tid * 4 + g]);
    fm[tid] = mm; fi[tid] = im; fz[tid] = Zm;
    plps[tid] = Wm / Zm - __logf(Zm);     // sum_k p*log p for this row
  }
  __syncthreads();
  if (tid == 0) {
    float s = 0.0f;
    for (int r2 = 0; r2 < MB; ++r2) s += plps[r2];
    atomicAdd(&ws[1024], s);
  }

  // -------- pass 2: recompute tiles, accumulate avg-prob column sums --------
  float pfm[8], prz[8];
#pragma unroll
  for (int r = 0; r < 8; ++r) {
    pfm[r] = fm[rslot + r];
    prz[r] = 1.0f / fz[rslot + r];
  }

  for (int t = 0; t < 16; ++t) {
    const int ct = cg + 4 * t;
    const int col = ct * 16 + ln;
    const float* bp = &cb[(size_t)col * D + 2 * h];
    const float* ap = &xs[(rt * 16 + ln) * XS_STRIDE + 2 * h];
    v8f acc = {};
#pragma unroll 8
    for (int k0 = 0; k0 < D; k0 += 4) {
      v2f a = *reinterpret_cast<const v2f*>(ap + k0);
      v2f b = *reinterpret_cast<const v2f*>(bp + k0);
      acc = __builtin_amdgcn_wmma_f32_16x16x4_f32(false, a, false, b,
                                                  (short)0, acc, false, false);
    }
    const float c2v = c2s[col];
    float ps = 0.0f;
#pragma unroll
    for (int r = 0; r < 8; ++r) {
      float d = x2v[r] + c2v - 2.0f * acc[r];
      ps += __expf(-100.0f * (d - pfm[r])) * prz[r];
    }
    atomicAdd(&colsum[col], ps);   // ds_add_f32
  }
  __syncthreads();
  for (int k = tid; k < K; k += 256) atomicAdd(&ws[k], colsum[k]);

  // -------- phase 3: gather quantized rows, write outputs, MSE --------
  const size_t ND = (size_t)N * D;
  float lsq = 0.0f;
#pragma unroll 4
  for (int it = 0; it < MB; ++it) {
    int f = it * 256 + tid;
    int row = f >> 8;
    int c   = f & 255;
    float q    = cb[(size_t)fi[row] * D + c];
    float xv   = xs[row * XS_STRIDE + c];
    float diff = q - xv;
    out[(size_t)(rowbase + row) * D + c] = xv + diff;  // straight-through value
    lsq += diff * diff;
  }
  atomicAdd(&s_red, lsq);
  __syncthreads();
  if (tid == 0) atomicAdd(&ws[1025], s_red);
  if (tid < MB) out[ND + 1 + rowbase + tid] = (float)fi[tid];
}

__global__ __launch_bounds__(256) void vq_final(float* __restrict__ out,
                                                const float* __restrict__ ws,
                                                int N) {
  __shared__ float red[256];
  int t = threadIdx.x;
  float invN = 1.0f / (float)N;
  float ae = 0.0f;
  for (int k = t; k < K; k += 256) {
    float a = ws[k] * invN;                 // avg_probs[k]
    ae += a * __logf(a + 1e-5f);
  }
  red[t] = ae;
  __syncthreads();
  for (int s = 128; s > 0; s >>= 1) {
    if (t < s) red[t] += red[t + s];
    __syncthreads();
  }
  if (t == 0) {
    float avg_entropy    = -red[0];
    float sample_entropy = -ws[1024] * invN;
    float mse            = ws[1025] / ((float)N * (float)D);
    // e_latent(0.25) + q_latent(1.0) share the same value -> 1.25 * mse
    float loss = 1.25f * mse + 0.1f * (sample_entropy - avg_entropy);
    out[(size_t)N * D] = loss;
  }
}

extern "C" void kernel_launch(void* const* d_in, const int* in_sizes, int n_in,
                              void* d_out, int out_size, void* d_ws, size_t ws_size,
                              hipStream_t stream) {
  (void)n_in; (void)out_size; (void)ws_size;
  const float* x  = (const float*)d_in[0];
  const float* cb = (const float*)d_in[1];
  float* out = (float*)d_out;
  float* ws  = (float*)d_ws;
  const int N = in_sizes[0] / D;   // 32768

  vq_init <<<K / 256, 256, 0, stream>>>(cb, ws);
  vq_main <<<N / MB, 256, 0, stream>>>(x, cb, out, ws, N);
  vq_final<<<1,      256, 0, stream>>>(out, ws, N);
}